// UVNetGraph_6760278524475
// MI455X (gfx1250) — compile-verified
//
#include <hip/hip_runtime.h>

// ---------------------------------------------------------------------------
// UVNet graph encoder for gfx1250 (MI455X), f32 WMMA (v_wmma_f32_16x16x4_f32).
// Edge messages computed as a fused WMMA GEMM with on-the-fly outer-product
// A-operand generation (avoids the 800MB/layer per-edge weight tensor).
// ---------------------------------------------------------------------------

typedef __attribute__((ext_vector_type(2))) float v2f;
typedef __attribute__((ext_vector_type(8))) float v8f;

#define NN 12500   // nodes
#define NE 50000   // edges

__device__ __forceinline__ v8f wmma_f32(v2f a, v2f b, v8f c) {
  // 8 args: (neg_a, A, neg_b, B, c_mod, C, reuse_a, reuse_b)
  return __builtin_amdgcn_wmma_f32_16x16x4_f32(false, a, false, b, (short)0, c,
                                               false, false);
}

__device__ __forceinline__ void atomAddF(float* p, float v) {
  unsafeAtomicAdd(p, v);   // -> global_atomic_add_f32
}

// ---------------------------------------------------------------------------
// Generic C[M,Ncols] = A[M,K](lda) @ B[K,Ncols](ldb) + bias[Ncols]
// One wave per 16x16 output tile. K must be a multiple of 4.
// grid = (Ncols/16, ceil(M/16)), block = 32.
// ---------------------------------------------------------------------------
__global__ __launch_bounds__(32)
void gemm_bias_wmma(const float* __restrict__ A, int lda,
                    const float* __restrict__ B, int ldb,
                    const float* __restrict__ bias,
                    float* __restrict__ C, int ldc, int M, int K) {
  const int lane = threadIdx.x;
  const int half = lane >> 4, l16 = lane & 15;
  const int tn = blockIdx.x * 16;
  const int tm = blockIdx.y * 16;
  int rowA = tm + l16;
  if (rowA >= M) rowA = M - 1;            // clamp; stores are masked below
  const float* Arow = A + (size_t)rowA * lda;
  v8f acc = {};
  for (int kk = 0; kk < K; kk += 4) {
    const int k0 = kk + 2 * half;         // lanes 0-15: K=kk,kk+1 ; 16-31: kk+2,kk+3
    v2f a = *reinterpret_cast<const v2f*>(Arow + k0);
    v2f b;
    b.x = B[(size_t)k0 * ldb + tn + l16];
    b.y = B[(size_t)(k0 + 1) * ldb + tn + l16];
    acc = wmma_f32(a, b, acc);
  }
  const float bv = bias ? bias[tn + l16] : 0.0f;
#pragma unroll
  for (int r = 0; r < 8; r++) {
    const int row = tm + r + (half << 3); // D: vgpr r = rows r / r+8
    if (row < M) C[(size_t)row * ldc + tn + l16] = acc[r] + bv;
  }
}

// ---------------------------------------------------------------------------
// NodeConv fused edge-message GEMM + scatter.
//   m[e,o] = sum_{j,i} ef[e,j]*hs[src,i]*Wf[(j*64+i)*64+o] + sum_i hs[src,i]*bf[i*64+o]
//   agg[dst[e],o] += m[e,o]
// hs = (1+eps)*h, nf hardcoded = 64. One wave per 16-edge tile, grid = NE/16.
// ---------------------------------------------------------------------------
__global__ __launch_bounds__(32)
void edge_msg_wmma(const float* __restrict__ h,
                   const float* __restrict__ ef, int lde, int efdim,
                   const int* __restrict__ src, const int* __restrict__ dst,
                   const float* __restrict__ Wf, const float* __restrict__ bf,
                   const float* __restrict__ epsp,
                   float* __restrict__ agg) {
  __shared__ float sh_h[16 * 64];
  __shared__ float sh_e[16 * 64];
  __shared__ int s_src[16], s_dst[16];
  const int lane = threadIdx.x;
  const int half = lane >> 4, l16 = lane & 15;
  const int e0 = blockIdx.x * 16;
  const float scale = 1.0f + epsp[0];
  if (lane < 16) {
    s_src[lane] = src[e0 + lane];
    s_dst[lane] = dst[e0 + lane];
  }
  __syncthreads();
  for (int i = lane; i < 16 * 64; i += 32) {
    const int t = i >> 6, c = i & 63;
    sh_h[i] = scale * h[(size_t)s_src[t] * 64 + c];
  }
  for (int i = lane; i < 16 * efdim; i += 32) {
    const int t = i / efdim, c = i - t * efdim;
    sh_e[i] = ef[(size_t)(e0 + t) * lde + c];
  }
  __syncthreads();

  const float* she = sh_e + l16 * efdim;
  const float* shh = sh_h + (l16 << 6);
  v8f a0 = {}, a1 = {}, a2 = {}, a3 = {};

  // main bilinear part: K = efdim*64, A generated as outer product from LDS
  const int KP = efdim << 6;
  for (int kk = 0; kk < KP; kk += 4) {
    const int k0 = kk + 2 * half;
    const int j = k0 >> 6, i0 = k0 & 63;   // k0 even -> k0,k0+1 share j
    const float ev = she[j];
    v2f a;
    a.x = ev * shh[i0];
    a.y = ev * shh[i0 + 1];
    const float* B0 = Wf + (size_t)k0 * 64 + l16;
    const float* B1 = B0 + 64;
    v2f b;
    b.x = B0[0];  b.y = B1[0];  a0 = wmma_f32(a, b, a0);
    b.x = B0[16]; b.y = B1[16]; a1 = wmma_f32(a, b, a1);
    b.x = B0[32]; b.y = B1[32]; a2 = wmma_f32(a, b, a2);
    b.x = B0[48]; b.y = B1[48]; a3 = wmma_f32(a, b, a3);
  }
  // bias part: 64 extra K rows, A = hs, B = bf reshaped [64,64]
  for (int kk = 0; kk < 64; kk += 4) {
    const int k0 = kk + 2 * half;
    v2f a;
    a.x = shh[k0];
    a.y = shh[k0 + 1];
    const float* B0 = bf + (size_t)k0 * 64 + l16;
    const float* B1 = B0 + 64;
    v2f b;
    b.x = B0[0];  b.y = B1[0];  a0 = wmma_f32(a, b, a0);
    b.x = B0[16]; b.y = B1[16]; a1 = wmma_f32(a, b, a1);
    b.x = B0[32]; b.y = B1[32]; a2 = wmma_f32(a, b, a2);
    b.x = B0[48]; b.y = B1[48]; a3 = wmma_f32(a, b, a3);
  }
  // scatter: agg[dst[e], :] += m[e, :]
#pragma unroll
  for (int r = 0; r < 8; r++) {
    const int t = r + (half << 3);
    float* out = agg + (size_t)s_dst[t] * 64 + l16;
    atomAddF(out,      a0[r]);
    atomAddF(out + 16, a1[r]);
    atomAddF(out + 32, a2[r]);
    atomAddF(out + 48, a3[r]);
  }
}

// ---------------------------------------------------------------------------
// EdgeConv input: X[e,c] = (1+eps)*ef[e,c] + p[src[e],c] + p[dst[e],c]
// ---------------------------------------------------------------------------
__global__ void edge_x_kernel(const float* __restrict__ ein, int lde,
                              const float* __restrict__ p, int C,
                              const int* __restrict__ src,
                              const int* __restrict__ dst,
                              const float* __restrict__ epsp,
                              float* __restrict__ X, int total) {
  const int i = blockIdx.x * blockDim.x + threadIdx.x;
  if (i >= total) return;
  const float scale = 1.0f + epsp[0];
  const int e = i / C, c = i - e * C;
  X[i] = scale * ein[(size_t)e * lde + c] +
         p[(size_t)src[e] * C + c] + p[(size_t)dst[e] * C + c];
}

// ---------------------------------------------------------------------------
// BatchNorm (training-mode batch stats, C = 64 columns)
// ---------------------------------------------------------------------------
__global__ void zero_kernel(float* __restrict__ p, int n) {
  const int i = blockIdx.x * blockDim.x + threadIdx.x;
  if (i < n) p[i] = 0.0f;
}

__global__ void colstat_partial(const float* __restrict__ X, int ld, int M,
                                float* __restrict__ sums) {  // sums[0:64]=Σx, [64:128]=Σx²
  const int c = threadIdx.x & 63;
  const int rg = threadIdx.x >> 6;   // 0..3
  float s = 0.f, s2 = 0.f;
  for (int r = blockIdx.x * 4 + rg; r < M; r += gridDim.x * 4) {
    const float v = X[(size_t)r * ld + c];   // coalesced across lanes
    s += v; s2 += v * v;
  }
  __shared__ float sh[512];
  sh[threadIdx.x] = s;
  sh[256 + threadIdx.x] = s2;
  __syncthreads();
  if (rg == 0) {
    for (int g = 1; g < 4; g++) { s += sh[g * 64 + c]; s2 += sh[256 + g * 64 + c]; }
    atomAddF(&sums[c], s);
    atomAddF(&sums[64 + c], s2);
  }
}

__global__ void colstat_final(const float* __restrict__ sums, int M,
                              float* __restrict__ mean, float* __restrict__ rstd) {
  const int c = threadIdx.x;  // 64 threads
  const float mu = sums[c] / (float)M;
  const float var = sums[64 + c] / (float)M - mu * mu;
  mean[c] = mu;
  rstd[c] = rsqrtf(fmaxf(var, 0.0f) + 1e-5f);
}

// mode: 0=none, 1=relu, 2=leaky_relu(0.01)
__global__ void bn_act_kernel(const float* __restrict__ X,
                              const float* __restrict__ mean,
                              const float* __restrict__ rstd,
                              float* __restrict__ Y, int total, int mode) {
  const int i = blockIdx.x * blockDim.x + threadIdx.x;
  if (i >= total) return;
  const int c = i & 63;
  float v = (X[i] - mean[c]) * rstd[c];
  if (mode == 1) v = fmaxf(v, 0.0f);
  else if (mode == 2) v = (v > 0.0f) ? v : 0.01f * v;
  Y[i] = v;
}

// ---------------------------------------------------------------------------
// Host orchestration
// ---------------------------------------------------------------------------
extern "C" void kernel_launch(void* const* d_in, const int* in_sizes, int n_in,
                              void* d_out, int out_size, void* d_ws, size_t ws_size,
                              hipStream_t stream) {
  (void)n_in; (void)out_size; (void)ws_size;

  // -------- input index maps (dict insertion order OR jax sorted-key order)
  const bool sorted = (in_sizes[0] == NE);  // sorted flatten puts 'dst' first
  int i_h, i_ef, i_src, i_dst, l0, l1, iWo1, ibo1, iWo2, ibo2;
  int f_ncWf, f_ncbf, f_nceps, f_ncW1, f_ncb1, f_ncW2, f_ncb2;
  int f_ecWp, f_ecbp, f_eceps, f_ecW1, f_ecb1, f_ecW2, f_ecb2;
  if (!sorted) {
    i_h = 0; i_ef = 1; l0 = 2; l1 = 16;
    iWo1 = 30; ibo1 = 31; iWo2 = 32; ibo2 = 33; i_src = 34; i_dst = 35;
    f_ncWf = 0; f_ncbf = 1; f_nceps = 2; f_ncW1 = 3; f_ncb1 = 4; f_ncW2 = 5; f_ncb2 = 6;
    f_ecWp = 7; f_ecbp = 8; f_eceps = 9; f_ecW1 = 10; f_ecb1 = 11; f_ecW2 = 12; f_ecb2 = 13;
  } else {
    i_dst = 0; i_ef = 1; i_h = 2;
    iWo1 = 3; iWo2 = 4; ibo1 = 5; ibo2 = 6; l0 = 7; l1 = 21; i_src = 35;
    f_ecW1 = 0; f_ecW2 = 1; f_ecWp = 2; f_ecb1 = 3; f_ecb2 = 4; f_ecbp = 5; f_eceps = 6;
    f_ncW1 = 7; f_ncW2 = 8; f_ncWf = 9; f_ncb1 = 10; f_ncb2 = 11; f_ncbf = 12; f_nceps = 13;
  }
  auto F = [&](int i) { return (const float*)d_in[i]; };

  struct Layer {
    const float *Wf, *bf, *epsn, *nW1, *nb1, *nW2, *nb2;
    const float *Wp, *bp, *epse, *eW1, *eb1, *eW2, *eb2;
  };
  Layer L[2];
  const int base[2] = {l0, l1};
  for (int l = 0; l < 2; l++) {
    const int b = base[l];
    L[l].Wf = F(b + f_ncWf);  L[l].bf = F(b + f_ncbf);  L[l].epsn = F(b + f_nceps);
    L[l].nW1 = F(b + f_ncW1); L[l].nb1 = F(b + f_ncb1);
    L[l].nW2 = F(b + f_ncW2); L[l].nb2 = F(b + f_ncb2);
    L[l].Wp = F(b + f_ecWp);  L[l].bp = F(b + f_ecbp);  L[l].epse = F(b + f_eceps);
    L[l].eW1 = F(b + f_ecW1); L[l].eb1 = F(b + f_ecb1);
    L[l].eW2 = F(b + f_ecW2); L[l].eb2 = F(b + f_ecb2);
  }
  const float* h_in  = F(i_h);
  const float* ef_in = F(i_ef);
  const int* src = (const int*)d_in[i_src];
  const int* dst = (const int*)d_in[i_dst];
  float* out = (float*)d_out;

  // -------- workspace layout (floats), ~80 MB total
  float* ws  = (float*)d_ws;
  float* hB  = ws;                  // [N,64]  h after layer 0
  float* hC  = hB + NN * 64;        // [N,64]  h after layer 1
  float* eB  = hC + NN * 64;        // [E,64]  efeat after layer 0
  float* eC  = eB + (size_t)NE * 64;// [E,64]  efeat after layer 1
  float* agg = eC + (size_t)NE * 64;// [N,64]  NodeConv aggregation
  float* nT1 = agg + NN * 64;       // [N,64]  node temp (also EdgeConv proj p)
  float* nT2 = nT1 + NN * 64;       // [N,64]  node temp
  float* eX  = nT2 + NN * 64;       // [E,64]  edge temp
  float* eT1 = eX + (size_t)NE * 64;
  float* eT2 = eT1 + (size_t)NE * 64;
  float* stat = eT2 + (size_t)NE * 64;  // 256 floats: sums[128], mean[64], rstd[64]

  auto zero = [&](float* p, int n) {
    zero_kernel<<<dim3((n + 255) / 256), dim3(256), 0, stream>>>(p, n);
  };
  auto gemm = [&](const float* A, int lda, const float* B, int ldb,
                  const float* bias, float* C, int ldc, int M, int Ncols, int K) {
    gemm_bias_wmma<<<dim3(Ncols / 16, (M + 15) / 16), dim3(32), 0, stream>>>(
        A, lda, B, ldb, bias, C, ldc, M, K);
  };
  auto bn = [&](const float* X, float* Y, int M, int mode) {
    zero(stat, 128);
    colstat_partial<<<dim3(64), dim3(256), 0, stream>>>(X, 64, M, stat);
    colstat_final<<<dim3(1), dim3(64), 0, stream>>>(stat, M, stat + 128, stat + 192);
    bn_act_kernel<<<dim3((M * 64 + 255) / 256), dim3(256), 0, stream>>>(
        X, stat + 128, stat + 192, Y, M * 64, mode);
  };

  auto node_conv = [&](const Layer& P, const float* h, const float* e, int lde,
                       int efd, float* hout) {
    zero(agg, NN * 64);
    edge_msg_wmma<<<dim3(NE / 16), dim3(32), 0, stream>>>(
        h, e, lde, efd, src, dst, P.Wf, P.bf, P.epsn, agg);
    gemm(agg, 64, P.nW1, 64, P.nb1, nT1, 64, NN, 64, 64);
    bn(nT1, nT2, NN, 1);                       // BN + relu
    gemm(nT2, 64, P.nW2, 64, P.nb2, nT1, 64, NN, 64, 64);
    bn(nT1, hout, NN, 2);                      // BN + leaky_relu
  };

  auto edge_conv = [&](const Layer& P, const float* h, const float* e, int lde,
                       int efd, float* eout) {
    gemm(h, 64, P.Wp, efd, P.bp, nT1, efd, NN, efd, 64);       // p = h @ Wp + bp
    edge_x_kernel<<<dim3((NE * efd + 255) / 256), dim3(256), 0, stream>>>(
        e, lde, nT1, efd, src, dst, P.epse, eX, NE * efd);
    gemm(eX, efd, P.eW1, 64, P.eb1, eT1, 64, NE, 64, efd);
    bn(eT1, eT2, NE, 1);                       // BN + relu
    gemm(eT2, 64, P.eW2, 64, P.eb2, eT1, 64, NE, 64, 64);
    bn(eT1, eout, NE, 2);                      // BN + leaky_relu
  };

  // -------- layer 0 (node dim 64, edge dim 32)
  node_conv(L[0], h_in, ef_in, 32, 32, hB);
  edge_conv(L[0], hB, ef_in, 32, 32, eB);
  // -------- layer 1 (node dim 64, edge dim 64)
  node_conv(L[1], hB, eB, 64, 64, hC);
  edge_conv(L[1], hC, eB, 64, 64, eC);

  // -------- output embeddings (written straight to d_out)
  gemm(hC, 64, F(iWo1), 128, F(ibo1), out, 128, NN, 128, 64);
  gemm(eC, 64, F(iWo2), 128, F(ibo2), out + (size_t)NN * 128, 128, NE, 128, 64);
}